// CRFClassifier_18511309046358
// MI455X (gfx1250) — compile-verified
//
#include <hip/hip_runtime.h>
#include <hip/hip_bf16.h>
#include <math.h>
#include <stdint.h>

// ---------------------------------------------------------------------------
// CRF classifier for MI455X (gfx1250, wave32).
//   B=64, S=512, H=1024, L=64
//   out[0]            = loss (float)
//   out[1 .. 1+B*S-1] = predicted labels (as float)
// ---------------------------------------------------------------------------

#define CRF_B 64
#define CRF_S 512
#define CRF_H 1024
#define CRF_L 64

typedef __attribute__((ext_vector_type(2))) float v2f;
typedef __attribute__((ext_vector_type(8))) float v8f;
typedef unsigned int u32x4 __attribute__((ext_vector_type(4)));
typedef int          i32x4 __attribute__((ext_vector_type(4)));
typedef int          i32x8 __attribute__((ext_vector_type(8)));

// GEMM tiling: K is consumed in chunks of 64, double-buffered in LDS via TDM.
#define KCHUNK   64
#define NCHUNK   (CRF_H / KCHUNK)       // 16
#define WSTRIDE  72                     // 64 cols + 8 DWORD pad (bank-conflict free)
#define BUFBYTES (KCHUNK * WSTRIDE * 4) // 18432 B per buffer

// ---------------------------------------------------------------------------
// TDM: load one K-chunk of W ([KCHUNK rows x 64 cols] f32, row-major) into LDS
// with an 8-DWORD pad after every 64-DWORD row (row stride = 72 words).
// Descriptor per CDNA5 ISA.
// ---------------------------------------------------------------------------
__device__ __forceinline__ void tdm_load_w_chunk(const float* gsrc,
                                                 unsigned lds_byte_addr)
{
    const unsigned long long ga = (unsigned long long)(uintptr_t)gsrc;

    u32x4 g0;
    g0[0] = 1u;                                   // count=1, user descriptor
    g0[1] = lds_byte_addr;                        // LDS destination (bytes)
    g0[2] = (unsigned)(ga & 0xFFFFFFFFu);         // global_addr[31:0]
    g0[3] = (unsigned)((ga >> 32) & 0x01FFFFFFu)  // global_addr[56:32]
          | (2u << 30);                           // type = 2 ("image")

    i32x8 g1;
    g1[0] = (2 << 16)     // data_size = 4 bytes
          | (1 << 20)     // pad_enable
          | (5 << 22)     // pad_interval: every 64 DWORDs
          | (7 << 25);    // pad_amount: 8 DWORDs
    g1[1] = (64 << 16);   // tensor_dim0 = 64 (bits [79:48], low half here)
    g1[2] = (1024 << 16); // tensor_dim0 hi = 0 | tensor_dim1 = 1024 (low half)
    g1[3] = (64 << 16);   // tensor_dim1 hi = 0 | tile_dim0 = 64
    g1[4] = KCHUNK;       // tile_dim1 = KCHUNK | tile_dim2 = 0
    g1[5] = 64;           // tensor_dim0_stride[31:0] = 64 elements
    g1[6] = 0;            // stride hi | tensor_dim1_stride lo (unused, 2D)
    g1[7] = 0;

    i32x4 gz = {0, 0, 0, 0};
#if defined(__clang_major__) && (__clang_major__ >= 23)
    i32x8 gz8 = {0, 0, 0, 0, 0, 0, 0, 0};
    __builtin_amdgcn_tensor_load_to_lds(g0, g1, gz, gz, gz8, 0);
#else
    __builtin_amdgcn_tensor_load_to_lds(g0, g1, gz, gz, 0);
#endif
}

// ---------------------------------------------------------------------------
// Kernel 1: emissions = hidden @ W + b   using V_WMMA_F32_16X16X4_F32
//   M = B*S = 32768, N = 64, K = 1024.
//   Block = 128 threads (4 waves); wave w owns rows [(4*blk+w)*16, +16) and
//   all four 16-wide N tiles (A fragment reused 4x per K step).
//   W is staged into LDS by the Tensor Data Mover, double-buffered per K-chunk,
//   with descriptor-driven padding so B-fragment ds reads are conflict-free.
// ---------------------------------------------------------------------------
__global__ __launch_bounds__(128) void emis_gemm_wmma(
    const float* __restrict__ A,     // [M, H] hidden_states
    const float* __restrict__ W,     // [H, L]
    const float* __restrict__ bias,  // [L]
    float* __restrict__ C)           // [M, L] emissions
{
    __shared__ float wbuf[2][KCHUNK * WSTRIDE];   // 2 x 18 KB, TDM-filled

    const int lane = threadIdx.x & 31;
    const int wave = threadIdx.x >> 5;
    const int m0   = (blockIdx.x * 4 + wave) * 16;

    const int rlo  = lane & 15;        // row-in-tile / col-in-tile
    const int hi   = lane >> 4;        // 0: lanes 0-15, 1: lanes 16-31
    const int koff = hi << 1;          // lower lanes hold K{0,1}, upper K{2,3}

    const float* __restrict__ arow = A + (size_t)(m0 + rlo) * CRF_H;
    const unsigned ldsbase = (unsigned)(uintptr_t)(&wbuf[0][0]);

    v8f acc0 = {}; v8f acc1 = {}; v8f acc2 = {}; v8f acc3 = {};

    // prime the pipeline: chunk 0 -> buffer 0
    if (threadIdx.x < 32) {
        tdm_load_w_chunk(W, ldsbase);
        __builtin_amdgcn_s_wait_tensorcnt(0);
    }
    __syncthreads();

    for (int c = 0; c < NCHUNK; ++c) {
        const float* __restrict__ bbuf = &wbuf[c & 1][0];

        // async prefetch of the next W chunk into the other buffer
        if ((c + 1 < NCHUNK) && (threadIdx.x < 32))
            tdm_load_w_chunk(W + (size_t)(c + 1) * KCHUNK * CRF_L,
                             ldsbase + (unsigned)(((c + 1) & 1) * BUFBYTES));

        // prefetch next A chunk for this row (cacheline-granular)
        if (c + 1 < NCHUNK)
            __builtin_prefetch(arow + (c + 1) * KCHUNK, 0, 0);

        const int k0 = c * KCHUNK;
#pragma unroll 4
        for (int kk = 0; kk < KCHUNK; kk += 4) {
            const int kb = k0 + kk + koff;
            // A fragment: lane holds A[row, kb], A[row, kb+1] (8B aligned)
            v2f a = *(const v2f*)(arow + kb);

            // B fragments from padded LDS: rows kloc, kloc+1; banks disjoint
            const int kloc = kk + koff;
            const float* __restrict__ bp = bbuf + kloc * WSTRIDE + rlo;
            v2f b0 = { bp[ 0], bp[WSTRIDE     ] };
            v2f b1 = { bp[16], bp[WSTRIDE + 16] };
            v2f b2 = { bp[32], bp[WSTRIDE + 32] };
            v2f b3 = { bp[48], bp[WSTRIDE + 48] };

            acc0 = __builtin_amdgcn_wmma_f32_16x16x4_f32(false, a, false, b0,
                                                         (short)0, acc0, false, false);
            acc1 = __builtin_amdgcn_wmma_f32_16x16x4_f32(false, a, false, b1,
                                                         (short)0, acc1, false, false);
            acc2 = __builtin_amdgcn_wmma_f32_16x16x4_f32(false, a, false, b2,
                                                         (short)0, acc2, false, false);
            acc3 = __builtin_amdgcn_wmma_f32_16x16x4_f32(false, a, false, b3,
                                                         (short)0, acc3, false, false);
        }

        // make sure the prefetched chunk has landed, then swap buffers
        if (threadIdx.x < 32)
            __builtin_amdgcn_s_wait_tensorcnt(0);
        __syncthreads();
    }

    // D layout: VGPR r -> M = r (lanes 0-15) or r+8 (lanes 16-31), N = lane&15.
    const int mrow_base = m0 + hi * 8;
#pragma unroll
    for (int r = 0; r < 8; ++r) {
        float* crow = C + (size_t)(mrow_base + r) * CRF_L;
        crow[rlo +  0] = acc0[r] + bias[rlo +  0];
        crow[rlo + 16] = acc1[r] + bias[rlo + 16];
        crow[rlo + 32] = acc2[r] + bias[rlo + 32];
        crow[rlo + 48] = acc3[r] + bias[rlo + 48];
    }
}

// ---------------------------------------------------------------------------
// Kernel 2: gold-path CRF score per batch element (1 block, 64 threads).
// ---------------------------------------------------------------------------
__global__ __launch_bounds__(64) void crf_score_kernel(
    const float* __restrict__ emis,      // [B,S,L]
    const uint8_t* __restrict__ mask,    // [B,S] bool
    const int* __restrict__ labels,      // [B,S]
    const float* __restrict__ startT,    // [L]
    const float* __restrict__ endT,      // [L]
    const float* __restrict__ trans,     // [L,L]
    float* __restrict__ score_ws)        // [B]
{
    const int b = threadIdx.x;
    if (b >= CRF_B) return;

    const int t0 = labels[b * CRF_S];
    float sc = startT[t0] + emis[((size_t)b * CRF_S) * CRF_L + t0];
    int prev = t0;
    int seqlen = mask[b * CRF_S] ? 1 : 0;

    for (int s = 1; s < CRF_S; ++s) {
        const int t = labels[b * CRF_S + s];
        const float m = mask[b * CRF_S + s] ? 1.0f : 0.0f;
        sc += (trans[prev * CRF_L + t] +
               emis[((size_t)b * CRF_S + s) * CRF_L + t]) * m;
        seqlen += mask[b * CRF_S + s] ? 1 : 0;
        prev = t;
    }
    const int last = labels[b * CRF_S + (seqlen > 0 ? seqlen - 1 : 0)];
    score_ws[b] = sc + endT[last];
}

// ---------------------------------------------------------------------------
// Kernel 3: fused forward scan (logsumexp normalizer + Viterbi) + backtrack.
//   One 64-thread workgroup per batch element. transitions + full backpointer
//   history live in LDS (16KB + 32KB), backtrack runs from LDS.
// ---------------------------------------------------------------------------
__global__ __launch_bounds__(64) void crf_scan_kernel(
    const float* __restrict__ emis,      // [B,S,L]
    const uint8_t* __restrict__ mask,    // [B,S]
    const float* __restrict__ startT,
    const float* __restrict__ endT,
    const float* __restrict__ trans,     // [L,L]
    const float* __restrict__ score_ws,  // [B] (from kernel 2)
    const int* __restrict__ pad_id_ptr,  // scalar
    float* __restrict__ llh_ws,          // [B]
    float* __restrict__ out_labels)      // [B,S] (float-encoded tags)
{
    __shared__ float tr[CRF_L * CRF_L];          // 16 KB, row-major [i][j]
    __shared__ float alpha[CRF_L];
    __shared__ float vs[CRF_L];
    __shared__ unsigned char hist[CRF_S][CRF_L]; // 32 KB backpointers

    const int j = threadIdx.x;
    const int b = blockIdx.x;

    // stage transitions: thread j reads column j in the hot loop -> bank j
    for (int i = j; i < CRF_L * CRF_L; i += CRF_L) tr[i] = trans[i];

    const float a0 = startT[j] + emis[((size_t)b * CRF_S) * CRF_L + j];
    alpha[j] = a0;
    vs[j]    = a0;
    __syncthreads();

    for (int s = 1; s < CRF_S; ++s) {
        const float e = emis[((size_t)b * CRF_S + s) * CRF_L + j];
        const int   m = mask[b * CRF_S + s];   // uniform across the block

        // pass 1: shared max for logsumexp + Viterbi max/argmax
        float mx = -INFINITY, vb = -INFINITY;
        int   va = 0;
#pragma unroll 4
        for (int i = 0; i < CRF_L; ++i) {
            const float t  = tr[i * CRF_L + j];
            const float av = alpha[i] + t;
            mx = fmaxf(mx, av);
            const float vv = vs[i] + t;
            if (vv > vb) { vb = vv; va = i; }
        }
        // pass 2: stabilized sum of exponentials
        float sum = 0.0f;
#pragma unroll 4
        for (int i = 0; i < CRF_L; ++i)
            sum += __expf(alpha[i] + tr[i * CRF_L + j] - mx);

        const float na = mx + __logf(sum) + e;
        const float nv = vb + e;
        hist[s][j] = (unsigned char)va;

        __syncthreads();
        if (m) { alpha[j] = na; vs[j] = nv; }
        __syncthreads();
    }

    // final reductions + backtrack (serial in thread 0: runs once, 64 items)
    if (j == 0) {
        float mx = -INFINITY;
        for (int i = 0; i < CRF_L; ++i) mx = fmaxf(mx, alpha[i] + endT[i]);
        float sm = 0.0f;
        for (int i = 0; i < CRF_L; ++i) sm += __expf(alpha[i] + endT[i] - mx);
        const float logZ = mx + __logf(sm);
        llh_ws[b] = score_ws[b] - logZ;

        float bm = -INFINITY; int bt = 0;
        for (int i = 0; i < CRF_L; ++i) {
            const float v = vs[i] + endT[i];
            if (v > bm) { bm = v; bt = i; }
        }

        const int pad = *pad_id_ptr;
        int cur = bt;
        out_labels[b * CRF_S + (CRF_S - 1)] =
            mask[b * CRF_S + (CRF_S - 1)] ? (float)bt : (float)pad;
        for (int s = CRF_S - 1; s >= 1; --s) {
            if (mask[b * CRF_S + s]) cur = hist[s][cur];
            out_labels[b * CRF_S + (s - 1)] =
                mask[b * CRF_S + (s - 1)] ? (float)cur : (float)pad;
        }
    }
}

// ---------------------------------------------------------------------------
// Kernel 4: loss = -sum_b llh[b]  (deterministic serial reduce, 64 values)
// ---------------------------------------------------------------------------
__global__ void crf_loss_kernel(const float* __restrict__ llh, float* __restrict__ out)
{
    if (threadIdx.x == 0 && blockIdx.x == 0) {
        float s = 0.0f;
        for (int b = 0; b < CRF_B; ++b) s += llh[b];
        out[0] = -s;
    }
}

// ---------------------------------------------------------------------------
// Host launcher
// ---------------------------------------------------------------------------
extern "C" void kernel_launch(void* const* d_in, const int* in_sizes, int n_in,
                              void* d_out, int out_size, void* d_ws, size_t ws_size,
                              hipStream_t stream) {
    const float*   hidden = (const float*)d_in[0];    // [B,S,H]
    const uint8_t* mask   = (const uint8_t*)d_in[1];  // [B,S] bool (1 byte)
    const int*     labels = (const int*)d_in[2];      // [B,S]
    const float*   W      = (const float*)d_in[3];    // [H,L]
    const float*   bias   = (const float*)d_in[4];    // [L]
    const float*   startT = (const float*)d_in[5];    // [L]
    const float*   endT   = (const float*)d_in[6];    // [L]
    const float*   trans  = (const float*)d_in[7];    // [L,L]
    const int*     pad_id = (const int*)d_in[8];      // scalar

    float* out = (float*)d_out;             // [0] loss, [1..] labels

    // workspace layout
    float* emis  = (float*)d_ws;                          // B*S*L floats (8MB)
    float* score = emis + (size_t)CRF_B * CRF_S * CRF_L;  // [B]
    float* llh   = score + CRF_B;                         // [B]

    // 1) emissions GEMM (WMMA f32 + TDM-staged W): 512 blocks x 128 threads
    emis_gemm_wmma<<<(CRF_B * CRF_S) / 64, 128, 0, stream>>>(hidden, W, bias, emis);

    // 2) gold-path score
    crf_score_kernel<<<1, 64, 0, stream>>>(emis, mask, labels, startT, endT, trans, score);

    // 3) fused normalizer + Viterbi scan, one workgroup per batch element
    crf_scan_kernel<<<CRF_B, 64, 0, stream>>>(emis, mask, startT, endT, trans,
                                              score, pad_id, llh, out + 1);

    // 4) loss
    crf_loss_kernel<<<1, 64, 0, stream>>>(llh, out);
}